// ZambaBlock_69475390980169
// MI455X (gfx1250) — compile-verified
//
#include <hip/hip_runtime.h>

// ---------------- model constants ----------------
#define BATCH   2
#define SEQ     2048
#define DMODEL  1024
#define DINNER  2048
#define DSTATE  64
#define PW      (DINNER + 2*DSTATE)   // 2176
#define NHEADS  16
#define HD      64
#define NCHUNK  (SEQ/128)             // 16

typedef float v2f __attribute__((ext_vector_type(2)));
typedef float v8f __attribute__((ext_vector_type(8)));

__device__ __forceinline__ v8f wmma4(v2f a, v2f b, v8f c) {
  // v_wmma_f32_16x16x4_f32 : D = A(16x4) x B(4x16) + C(16x16), all f32
  return __builtin_amdgcn_wmma_f32_16x16x4_f32(false, a, false, b, (short)0, c, false, false);
}

// async memory->LDS copy, 16B per lane; tracked by ASYNCcnt
__device__ __forceinline__ void async_b128(void* lds, const void* g) {
  unsigned off = (unsigned)(unsigned long long)lds;   // low 32 bits = LDS byte offset
  asm volatile("global_load_async_to_lds_b128 %0, %1, off"
               :: "v"(off), "v"(g) : "memory");
}
__device__ __forceinline__ void wait_async0() {
  asm volatile("s_wait_asynccnt 0x0" ::: "memory");
}

// =====================================================================
// FAST GEMM: C[z] = A[z] @ B[z] (+C if beta). Requires:
//   A,B row-major (cs==1), M%128==0, N%64==0, K%16==0, 16B-aligned bases,
//   rsA%4==0, rsB%4==0.
// Async B128 loads -> double-buffered LDS; 8 waves, 32x32 per wave.
// =====================================================================
__global__ __launch_bounds__(256)
void gemm_wmma_fast(const float* __restrict__ A, const float* __restrict__ B,
                    float* __restrict__ C,
                    int M, int N, int K,
                    long long rsA, long long rsB, long long ldC,
                    long long aZ, long long bZ, long long cZ,
                    int beta, int zmod)
{
  const int z = blockIdx.z;
  if (zmod > 0 && (z % zmod) == 0) return;
  A += (long long)z * aZ; B += (long long)z * bZ; C += (long long)z * cZ;

  const int bm = blockIdx.y * 128;
  const int bn = blockIdx.x * 64;

  __shared__ float As[2][128][20];   // row stride 80B (16B aligned), conflict-free
  __shared__ float Bs[2][16][68];    // row stride 272B

  const int tid  = threadIdx.x;
  const int wid  = tid >> 5, lane = tid & 31;
  const int half = lane >> 4, lq = lane & 15;
  const int wm   = (wid & 3) * 32, wn = (wid >> 2) * 32;

  // per-thread load coords (float4 granules)
  const int ra = tid >> 2, ca = (tid & 3) * 4;     // A: 2 rows (ra, ra+64) x 4 k
  const int rb = tid >> 4, cb = (tid & 15) * 4;    // B: 1 row x 4 cols

  auto issue = [&](int buf, int k0) {
    async_b128(&As[buf][ra][ca],      A + (long long)(bm + ra)      * rsA + k0 + ca);
    async_b128(&As[buf][ra + 64][ca], A + (long long)(bm + ra + 64) * rsA + k0 + ca);
    async_b128(&Bs[buf][rb][cb],      B + (long long)(k0 + rb)      * rsB + bn + cb);
  };

  v8f acc[2][2];
  const v8f z8 = {0,0,0,0,0,0,0,0};
  acc[0][0]=z8; acc[0][1]=z8; acc[1][0]=z8; acc[1][1]=z8;

  issue(0, 0);
  wait_async0();
  __syncthreads();

  int buf = 0;
  for (int k0 = 0; k0 < K; k0 += 16) {
    if (k0 + 16 < K) issue(buf ^ 1, k0 + 16);   // prefetch next tile (async DMA)
#pragma unroll
    for (int kk = 0; kk < 16; kk += 4) {
      int kb = kk + half*2;
      v2f a0, a1, b0, b1;
      a0[0] = As[buf][wm + lq][kb];        a0[1] = As[buf][wm + lq][kb+1];
      a1[0] = As[buf][wm + 16 + lq][kb];   a1[1] = As[buf][wm + 16 + lq][kb+1];
      b0[0] = Bs[buf][kb][wn + lq];        b0[1] = Bs[buf][kb+1][wn + lq];
      b1[0] = Bs[buf][kb][wn + 16 + lq];   b1[1] = Bs[buf][kb+1][wn + 16 + lq];
      acc[0][0] = wmma4(a0, b0, acc[0][0]);
      acc[0][1] = wmma4(a0, b1, acc[0][1]);
      acc[1][0] = wmma4(a1, b0, acc[1][0]);
      acc[1][1] = wmma4(a1, b1, acc[1][1]);
    }
    wait_async0();
    __syncthreads();
    buf ^= 1;
  }

#pragma unroll
  for (int mi = 0; mi < 2; mi++)
#pragma unroll
    for (int ni = 0; ni < 2; ni++)
#pragma unroll
      for (int r = 0; r < 8; r++) {
        int row = bm + wm + mi*16 + r + half*8;
        int col = bn + wn + ni*16 + lq;
        long long off = (long long)row*ldC + col;
        float v = acc[mi][ni][r];
        if (beta) v += C[off];
        C[off] = v;
      }
}

// =====================================================================
// GENERIC GEMM (strided/transposed/guarded); used for the two small
// transposed scan GEMMs. epi==1: decay mask C[i][j]*= (i>=j)?(-.5)^(i-j):0
// =====================================================================
__global__ __launch_bounds__(256)
void gemm_wmma(const float* __restrict__ A, const float* __restrict__ B,
               float* __restrict__ C,
               int M, int N, int K,
               long long rsA, long long csA, long long rsB, long long csB,
               long long ldC, long long aZ, long long bZ, long long cZ,
               int beta, int epi, int zmod)
{
  const int z = blockIdx.z;
  if (zmod > 0 && (z % zmod) == 0) return;
  A += (long long)z * aZ; B += (long long)z * bZ; C += (long long)z * cZ;

  const int bm = blockIdx.y * 128;
  const int bn = blockIdx.x * 64;

  __shared__ float As[128][17];
  __shared__ float Bs[16][65];

  const int tid  = threadIdx.x;
  const int wid  = tid >> 5, lane = tid & 31;
  const int half = lane >> 4, lq = lane & 15;
  const int wm   = (wid & 3) * 32, wn = (wid >> 2) * 32;

  v8f acc[2][2];
  const v8f z8 = {0,0,0,0,0,0,0,0};
  acc[0][0]=z8; acc[0][1]=z8; acc[1][0]=z8; acc[1][1]=z8;

  for (int k0 = 0; k0 < K; k0 += 16) {
#pragma unroll
    for (int i = 0; i < 8; i++) {
      int e = tid + 256*i; int r = e >> 4, c = e & 15;
      int gr = bm + r, gc = k0 + c;
      As[r][c] = (gr < M && gc < K) ? A[(long long)gr*rsA + (long long)gc*csA] : 0.f;
    }
#pragma unroll
    for (int i = 0; i < 4; i++) {
      int e = tid + 256*i; int kk = e >> 6, n = e & 63;
      int gk = k0 + kk, gn = bn + n;
      Bs[kk][n] = (gk < K && gn < N) ? B[(long long)gk*rsB + (long long)gn*csB] : 0.f;
    }
    __syncthreads();
#pragma unroll
    for (int kk = 0; kk < 16; kk += 4) {
      int kb = kk + half*2;
      v2f a0, a1, b0, b1;
      a0[0] = As[wm + lq][kb];        a0[1] = As[wm + lq][kb+1];
      a1[0] = As[wm + 16 + lq][kb];   a1[1] = As[wm + 16 + lq][kb+1];
      b0[0] = Bs[kb][wn + lq];        b0[1] = Bs[kb+1][wn + lq];
      b1[0] = Bs[kb][wn + 16 + lq];   b1[1] = Bs[kb+1][wn + 16 + lq];
      acc[0][0] = wmma4(a0, b0, acc[0][0]);
      acc[0][1] = wmma4(a0, b1, acc[0][1]);
      acc[1][0] = wmma4(a1, b0, acc[1][0]);
      acc[1][1] = wmma4(a1, b1, acc[1][1]);
    }
    __syncthreads();
  }

#pragma unroll
  for (int mi = 0; mi < 2; mi++)
#pragma unroll
    for (int ni = 0; ni < 2; ni++)
#pragma unroll
      for (int r = 0; r < 8; r++) {
        int row = bm + wm + mi*16 + r + half*8;
        int col = bn + wn + ni*16 + lq;
        if (row < M && col < N) {
          float v = acc[mi][ni][r];
          if (epi == 1) {
            int d = row - col;
            v = (d >= 0) ? ldexpf((d & 1) ? -v : v, -d) : 0.f;
          }
          long long off = (long long)row*ldC + col;
          if (beta) v += C[off];
          C[off] = v;
        }
      }
}

// ---------------- depthwise causal conv(4) + SiLU ----------------
__global__ void conv_silu_kernel(const float* __restrict__ proj,
                                 const float* __restrict__ cw,
                                 const float* __restrict__ cb,
                                 float* __restrict__ xc)
{
  long long idx = (long long)blockIdx.x * 256 + threadIdx.x;
  if (idx >= (long long)BATCH*SEQ*DINNER) return;
  int d = (int)(idx % DINNER);
  long long bt = idx / DINNER;
  int t = (int)(bt % SEQ);
  long long b = bt / SEQ;
  float s = cb[d];
#pragma unroll
  for (int k = 0; k < 4; k++) {
    int tt = t - 3 + k;
    if (tt >= 0) s += cw[d*4 + k] * proj[((long long)(b*SEQ + tt))*PW + d];
  }
  xc[idx] = s / (1.f + __expf(-s));
}

// ---------------- B/C chunk extraction + decay-weighted copies ----------------
__global__ void prep_bc_kernel(const float* __restrict__ proj,
                               float* __restrict__ Bd, float* __restrict__ Cd,
                               float* __restrict__ Bw, float* __restrict__ Ca)
{
  int idx = blockIdx.x * 256 + threadIdx.x;
  if (idx >= BATCH*SEQ*DSTATE) return;
  int s  = idx & 63;
  int bt = idx >> 6;
  int t  = bt % SEQ, b = bt / SEQ;
  int j  = t & 127,  c = t >> 7;
  long long pr = ((long long)(b*SEQ + t))*PW + DINNER;
  float bm = proj[pr + s];
  float cm = proj[pr + DSTATE + s];
  long long o = ((long long)(b*NCHUNK + c))*128*64 + j*64 + s;
  Bd[o] = bm;  Cd[o] = cm;
  int e1 = 127 - j;                              // a^(127-j), a = -0.5
  Bw[o] = ldexpf((e1 & 1) ? -bm : bm, -e1);
  int e2 = j + 1;                                // a^(j+1)
  Ca[o] = ldexpf((e2 & 1) ? -cm : cm, -e2);
}

// ---------------- residual add + LayerNorm ----------------
__global__ __launch_bounds__(256)
void resid_ln_kernel(const float* __restrict__ ylin, const float* __restrict__ res,
                     const float* __restrict__ g, const float* __restrict__ bt,
                     float* __restrict__ out)
{
  long long row = blockIdx.x;
  const float* yr = ylin + row*DMODEL;
  const float* rr = res  + row*DMODEL;
  float* outr = out + row*DMODEL;
  int tid = threadIdx.x;
  float v[4]; float s = 0.f;
#pragma unroll
  for (int i = 0; i < 4; i++) { int c = tid + i*256; v[i] = yr[c] + rr[c]; s += v[i]; }
  __shared__ float red[8];
#pragma unroll
  for (int off = 16; off; off >>= 1) s += __shfl_xor(s, off, 32);
  if ((tid & 31) == 0) red[tid >> 5] = s;
  __syncthreads();
  float tot = 0.f;
#pragma unroll
  for (int w = 0; w < 8; w++) tot += red[w];
  float mu = tot * (1.f/1024.f);
  float s2 = 0.f;
#pragma unroll
  for (int i = 0; i < 4; i++) { float d = v[i]-mu; s2 += d*d; }
#pragma unroll
  for (int off = 16; off; off >>= 1) s2 += __shfl_xor(s2, off, 32);
  __syncthreads();
  if ((tid & 31) == 0) red[tid >> 5] = s2;
  __syncthreads();
  float tv = 0.f;
#pragma unroll
  for (int w = 0; w < 8; w++) tv += red[w];
  float rstd = rsqrtf(tv * (1.f/1024.f) + 1e-5f);
#pragma unroll
  for (int i = 0; i < 4; i++) { int c = tid + i*256; outr[c] = (v[i]-mu)*rstd*g[c] + bt[c]; }
}

__global__ void copy_f32(const float* __restrict__ in, float* __restrict__ out, long long n)
{
  long long i = (long long)blockIdx.x * 256 + threadIdx.x;
  if (i < n) out[i] = in[i];
}

// ---------------- flash attention (causal), WMMA f32, async K/V ----------------
__global__ __launch_bounds__(256)
void flash_attn_kernel(const float* __restrict__ qb, const float* __restrict__ kb,
                       const float* __restrict__ vb, float* __restrict__ ob)
{
  const int qt = blockIdx.x, h = blockIdx.y, b = blockIdx.z;
  __shared__ float Qs[128][68];
  __shared__ float Ks[2][16][68];
  __shared__ float Vs[2][16][68];
  __shared__ float Ps[8][16][17];
  const int tid = threadIdx.x;
  const int wid = tid >> 5, lane = tid & 31, half = lane >> 4, lq = lane & 15;
  const long long rs = (long long)NHEADS*HD;   // 1024

  const float* qp = qb + ((long long)(b*SEQ + qt*128))*rs + h*HD;
  const float* kp = kb + ((long long)(b*SEQ))*rs + h*HD;
  const float* vp = vb + ((long long)(b*SEQ))*rs + h*HD;

  const int rkv = tid >> 4, dkv = (tid & 15) * 4;
  auto issueKV = [&](int buf, int kt) {
    long long off = (long long)(kt*16 + rkv)*rs + dkv;
    async_b128(&Ks[buf][rkv][dkv], kp + off);
    async_b128(&Vs[buf][rkv][dkv], vp + off);
  };

#pragma unroll
  for (int i = 0; i < 8; i++) {                  // Q tile (raw; scale applied later)
    int e = tid + 256*i; int r = e >> 4, d4 = (e & 15) * 4;
    async_b128(&Qs[r][d4], qp + (long long)r*rs + d4);
  }
  issueKV(0, 0);
  wait_async0();
  __syncthreads();

  const v8f z8 = {0,0,0,0,0,0,0,0};
  v8f o[4]; o[0]=z8; o[1]=z8; o[2]=z8; o[3]=z8;
  float m[8], l[8];
#pragma unroll
  for (int r = 0; r < 8; r++) { m[r] = -1e30f; l[r] = 0.f; }

  const int q0  = qt*128 + wid*16;
  const int nkt = qt*8 + 8;
  int buf = 0;

  for (int kt = 0; kt < nkt; kt++) {
    if (kt + 1 < nkt) issueKV(buf ^ 1, kt + 1);

    v8f s = z8;                                  // S = Q K^T  (16x16, K=64)
#pragma unroll
    for (int kk = 0; kk < HD; kk += 4) {
      int kbx = kk + half*2;
      v2f a, bbv;
      a[0]   = Qs[wid*16 + lq][kbx];       a[1]   = Qs[wid*16 + lq][kbx+1];
      bbv[0] = Ks[buf][lq][kbx];           bbv[1] = Ks[buf][lq][kbx+1];
      s = wmma4(a, bbv, s);
    }

    float fac[8];
#pragma unroll
    for (int r = 0; r < 8; r++) {
      int qrow = q0 + r + half*8;
      int kcol = kt*16 + lq;
      float sv = (kcol <= qrow) ? s[r] * 0.125f : -1e30f;
      float mx = sv;
#pragma unroll
      for (int off = 8; off; off >>= 1) mx = fmaxf(mx, __shfl_xor(mx, off, 16));
      float mnew = fmaxf(m[r], mx);
      fac[r] = __expf(m[r] - mnew);
      float pv = (sv <= -1e29f) ? 0.f : __expf(sv - mnew);
      float ps = pv;
#pragma unroll
      for (int off = 8; off; off >>= 1) ps += __shfl_xor(ps, off, 16);
      l[r] = l[r]*fac[r] + ps;
      m[r] = mnew;
      Ps[wid][r + half*8][lq] = pv;
    }
#pragma unroll
    for (int n = 0; n < 4; n++)
#pragma unroll
      for (int r = 0; r < 8; r++) o[n][r] *= fac[r];
    __syncthreads();                             // Ps visible cross-lane

#pragma unroll
    for (int n = 0; n < 4; n++) {                // O += P @ V
#pragma unroll
      for (int kk = 0; kk < 16; kk += 4) {
        int kbx = kk + half*2;
        v2f a, bbv;
        a[0]   = Ps[wid][lq][kbx];           a[1]   = Ps[wid][lq][kbx+1];
        bbv[0] = Vs[buf][kbx][n*16 + lq];    bbv[1] = Vs[buf][kbx+1][n*16 + lq];
        o[n] = wmma4(a, bbv, o[n]);
      }
    }
    wait_async0();
    __syncthreads();
    buf ^= 1;
  }

  float* op = ob + ((long long)(b*SEQ + qt*128))*rs + h*HD;
#pragma unroll
  for (int n = 0; n < 4; n++)
#pragma unroll
    for (int r = 0; r < 8; r++) {
      int qr = wid*16 + r + half*8;
      op[(long long)qr*rs + n*16 + lq] = o[n][r] / l[r];
    }
}

// =====================================================================
// host side
// =====================================================================
static inline void gemm_fast(hipStream_t st, const float* A, const float* B, float* C,
                             int M, int N, int K, long long rsA, long long rsB, long long ldC,
                             long long aZ, long long bZ, long long cZ, int Z, int beta, int zmod)
{
  dim3 g(N / 64, M / 128, Z);
  gemm_wmma_fast<<<g, dim3(256), 0, st>>>(A, B, C, M, N, K, rsA, rsB, ldC, aZ, bZ, cZ, beta, zmod);
}

static inline void gemm_gen(hipStream_t st, const float* A, const float* B, float* C,
                            int M, int N, int K,
                            long long rsA, long long csA, long long rsB, long long csB,
                            long long ldC, long long aZ, long long bZ, long long cZ,
                            int Z, int beta, int epi, int zmod)
{
  dim3 g((N + 63) / 64, (M + 127) / 128, Z);
  gemm_wmma<<<g, dim3(256), 0, st>>>(A, B, C, M, N, K, rsA, csA, rsB, csB,
                                     ldC, aZ, bZ, cZ, beta, epi, zmod);
}

extern "C" void kernel_launch(void* const* d_in, const int* in_sizes, int n_in,
                              void* d_out, int out_size, void* d_ws, size_t ws_size,
                              hipStream_t stream)
{
  const float* x_in  = (const float*)d_in[0];
  const float* Wi    = (const float*)d_in[1];   // (5,1024,2176)
  const float* cw    = (const float*)d_in[2];   // (5,2048,4)
  const float* cb    = (const float*)d_in[3];   // (5,2048)
  const float* Wo    = (const float*)d_in[4];   // (5,2048,1024)
  const float* lng   = (const float*)d_in[5];
  const float* lnb   = (const float*)d_in[6];
  const float* wq    = (const float*)d_in[7];
  const float* wk    = (const float*)d_in[8];
  const float* wv    = (const float*)d_in[9];
  const float* wo    = (const float*)d_in[10];
  const float* alng  = (const float*)d_in[11];
  const float* alnb  = (const float*)d_in[12];
  float* out = (float*)d_out;
  float* w = (float*)d_ws;

  const long long NTOK = (long long)BATCH*SEQ;          // 4096
  float* X    = w;
  float* PROJ = X    + NTOK*DMODEL;
  float* XC   = PROJ + NTOK*PW;
  float* Y    = XC   + NTOK*DINNER;
  float* YLIN = Y    + NTOK*DINNER;
  float* HT   = YLIN + NTOK*DMODEL;
  float* S    = HT   + (long long)BATCH*NCHUNK*DSTATE*DINNER;
  float* Bd   = S    + (long long)BATCH*NCHUNK*128*128;
  float* Cd   = Bd   + (long long)BATCH*SEQ*DSTATE;
  float* Bw   = Cd   + (long long)BATCH*SEQ*DSTATE;
  float* Ca   = Bw   + (long long)BATCH*SEQ*DSTATE;
  float* Qb = PROJ;                  // attention aliases (SSM scratch dead by layer 5)
  float* Kb = PROJ + NTOK*DMODEL;
  float* Vb = XC;
  float* Ob = Y;

  const long long CH_U  = 128ll*DINNER;
  const long long CH_H  = (long long)DSTATE*DINNER;
  const long long CH_S  = 128ll*128;
  const long long CH_BC = 128ll*DSTATE;
  const int Z = BATCH*NCHUNK;        // 32

  copy_f32<<<dim3((unsigned)((NTOK*DMODEL + 255)/256)), dim3(256), 0, stream>>>(x_in, X, NTOK*DMODEL);

  int ssm = 0;
  for (int layer = 0; layer < 6; layer++) {
    if (layer == 5) {
      gemm_fast(stream, X, wq, Qb, (int)NTOK, DMODEL, DMODEL, DMODEL, DMODEL, DMODEL, 0,0,0, 1, 0, 0);
      gemm_fast(stream, X, wk, Kb, (int)NTOK, DMODEL, DMODEL, DMODEL, DMODEL, DMODEL, 0,0,0, 1, 0, 0);
      gemm_fast(stream, X, wv, Vb, (int)NTOK, DMODEL, DMODEL, DMODEL, DMODEL, DMODEL, 0,0,0, 1, 0, 0);
      flash_attn_kernel<<<dim3(SEQ/128, NHEADS, BATCH), dim3(256), 0, stream>>>(Qb, Kb, Vb, Ob);
      gemm_fast(stream, Ob, wo, YLIN, (int)NTOK, DMODEL, DMODEL, DMODEL, DMODEL, DMODEL, 0,0,0, 1, 0, 0);
      resid_ln_kernel<<<dim3((unsigned)NTOK), dim3(256), 0, stream>>>(YLIN, X, alng, alnb, out);
    } else {
      const int j = ssm++;
      // 1) proj = X @ Wi[j]
      gemm_fast(stream, X, Wi + (long long)j*DMODEL*PW, PROJ,
                (int)NTOK, PW, DMODEL, DMODEL, PW, PW, 0,0,0, 1, 0, 0);
      // 2) xc = silu(causal depthwise conv)
      conv_silu_kernel<<<dim3((unsigned)((NTOK*DINNER + 255)/256)), dim3(256), 0, stream>>>(
          PROJ, cw + (long long)j*DINNER*4, cb + (long long)j*DINNER, XC);
      // 3) chunk B/C extraction + decay weights
      prep_bc_kernel<<<dim3((unsigned)((NTOK*DSTATE + 255)/256)), dim3(256), 0, stream>>>(
          PROJ, Bd, Cd, Bw, Ca);
      // 4) S[z] = (Cd @ Bd^T) * decay-mask      (128x128, K=64) — transposed B: generic
      gemm_gen(stream, Cd, Bd, S, 128, 128, DSTATE,
               DSTATE,1, 1,DSTATE, 128, CH_BC, CH_BC, CH_S, Z, 0, 1, 0);
      // 5) HT[z] = Bw^T @ U                     (64x2048, K=128) — transposed A: generic
      gemm_gen(stream, Bw, XC, HT, DSTATE, DINNER, 128,
               1,DSTATE, DINNER,1, DINNER, CH_BC, CH_U, CH_H, Z, 0, 0, 0);
      // 6) Y[z] = S @ U                         (128x2048, K=128)
      gemm_fast(stream, S, XC, Y, 128, DINNER, 128,
                128, DINNER, DINNER, CH_S, CH_U, CH_U, Z, 0, 0);
      // 7) Y[z] += Ca @ HT[z-1]   (skip chunk 0 of each batch)
      gemm_fast(stream, Ca, HT - CH_H, Y, 128, DINNER, DSTATE,
                DSTATE, DINNER, DINNER, CH_BC, CH_H, CH_U, Z, 1, NCHUNK);
      // 8) ylin = Y @ Wo[j]
      gemm_fast(stream, Y, Wo + (long long)j*DINNER*DMODEL, YLIN,
                (int)NTOK, DMODEL, DINNER, DINNER, DMODEL, DMODEL, 0,0,0, 1, 0, 0);
      // 9) X = LN(ylin + X)
      resid_ln_kernel<<<dim3((unsigned)NTOK), dim3(256), 0, stream>>>(
          YLIN, X, lng + (long long)j*DMODEL, lnb + (long long)j*DMODEL, X);
    }
  }
}